// SelectedLoss_6021544149734
// MI455X (gfx1250) — compile-verified
//
#include <hip/hip_runtime.h>

// ---------------------------------------------------------------------------
// SelectedLoss (dice over label-selected channels) for MI455X / gfx1250.
//
// loss = 1 - (1/(B*K*H)) * sum_{b,k,h} (2*inter + eps)/(sum_p + sum_t + eps)
// Bandwidth-bound: reads only the selected channels (67 MB) once with
// coalesced non-temporal B128 loads; ~2.9 us at 23.3 TB/s.
// Cross-lane reduction uses V_WMMA_F32_16X16X4_F32 with an all-ones B matrix
// (layout-invariant) + documented A / C-D f32 layouts.
// ---------------------------------------------------------------------------

typedef __attribute__((ext_vector_type(2))) float v2f;
typedef __attribute__((ext_vector_type(4))) float v4f;
typedef __attribute__((ext_vector_type(8))) float v8f;

#define EPSF 1e-6f

constexpr int B = 8, C = 16, H = 512, W = 512, K = 4;
constexpr int ROWS            = B * K * H;          // 16384 dice rows
constexpr int WAVES_PER_BLOCK = 8;                  // 256 threads = 8 wave32
constexpr int BLOCKS          = ROWS / WAVES_PER_BLOCK; // 2048
constexpr int F4_PER_ROW      = W / 4;              // 128 float4 per row

// Sum x across all 32 lanes using V_WMMA_F32_16X16X4_F32.
// A (16x4 f32, documented layout): lanes 0-15 hold {K0,K1}, lanes 16-31 {K2,K3}.
//   a = {x, 0}  =>  A[m][0]=x_m, A[m][1]=0, A[m][2]=x_{m+16}, A[m][3]=0.
// B = all ones (every entry 1.0 -> result independent of B's VGPR layout).
//   D[m][j] = x_m + x_{m+16}.
// C/D layout (documented): VGPR r, lane l<16 -> (M=r,   N=l),
//                                   lane l>=16 -> (M=8+r, N=l-16).
// Per-lane sum of the 8 D VGPRs covers rows 0-7 (lanes<16) / 8-15 (lanes>=16);
// one xor-16 shuffle-add yields the full 32-lane total in every lane.
__device__ __forceinline__ float wave_sum32(float x) {
  v2f a; a[0] = x;    a[1] = 0.0f;
  v2f b; b[0] = 1.0f; b[1] = 1.0f;
  v8f c = {};
  c = __builtin_amdgcn_wmma_f32_16x16x4_f32(
      /*neg_a=*/false, a, /*neg_b=*/false, b,
      /*c_mod=*/(short)0, c, /*reuse_a=*/false, /*reuse_b=*/false);
  float s = ((c[0] + c[1]) + (c[2] + c[3])) + ((c[4] + c[5]) + (c[6] + c[7]));
  s += __shfl_xor(s, 16, 32);
  return s;
}

__global__ void __launch_bounds__(256)
dice_rows_kernel(const float* __restrict__ pred,
                 const float* __restrict__ targ,
                 const int*   __restrict__ labelseq,
                 float*       __restrict__ partial) {
  __shared__ float wsum[WAVES_PER_BLOCK];

  const int tid  = threadIdx.x;
  const int wave = tid >> 5;
  const int lane = tid & 31;

  // One wave per (b,k,h) row.
  const int row = blockIdx.x * WAVES_PER_BLOCK + wave;   // 0..ROWS-1
  const int h   = row & (H - 1);
  const int bk  = row >> 9;                              // row / H
  const int b   = bk >> 2;                               // bk / K
  const int k   = bk & (K - 1);
  const int ch  = labelseq[b * K + k];                   // 0..C-1

  const size_t base = (((size_t)b * C + (size_t)ch) * H + (size_t)h) * (size_t)W;
  const v4f* __restrict__ p4 = reinterpret_cast<const v4f*>(pred + base); // rows are 2KB aligned
  const v4f* __restrict__ t4 = reinterpret_cast<const v4f*>(targ + base);

  float sp = 0.0f, st = 0.0f, si = 0.0f;
#pragma unroll
  for (int i = 0; i < F4_PER_ROW / 32; ++i) {            // 4 iterations
    const v4f pv = __builtin_nontemporal_load(&p4[lane + 32 * i]); // 512B/wave coalesced
    const v4f tv = __builtin_nontemporal_load(&t4[lane + 32 * i]);
    sp += (pv.x + pv.y) + (pv.z + pv.w);
    st += (tv.x + tv.y) + (tv.z + tv.w);
    si  = fmaf(pv.x, tv.x, si);
    si  = fmaf(pv.y, tv.y, si);
    si  = fmaf(pv.z, tv.z, si);
    si  = fmaf(pv.w, tv.w, si);
  }

  // union = sum_p + sum_t = wave_sum(sp + st): one WMMA instead of two.
  const float inter = wave_sum32(si);
  const float usum  = wave_sum32(sp + st);

  if (lane == 0)
    wsum[wave] = (2.0f * inter + EPSF) / (usum + EPSF);
  __syncthreads();

  if (tid == 0) {
    float acc = 0.0f;
#pragma unroll
    for (int w = 0; w < WAVES_PER_BLOCK; ++w) acc += wsum[w];  // fixed order
    partial[blockIdx.x] = acc;
  }
}

__global__ void __launch_bounds__(256)
finalize_kernel(const float* __restrict__ partial, float* __restrict__ out, int n) {
  __shared__ float sdata[256];
  const int tid = threadIdx.x;
  float s = 0.0f;
  for (int i = tid; i < n; i += 256) s += partial[i];            // fixed order
  sdata[tid] = s;
  __syncthreads();
  for (int stride = 128; stride > 0; stride >>= 1) {             // deterministic tree
    if (tid < stride) sdata[tid] += sdata[tid + stride];
    __syncthreads();
  }
  if (tid == 0)
    out[0] = 1.0f - sdata[0] * (1.0f / (float)ROWS);
}

extern "C" void kernel_launch(void* const* d_in, const int* in_sizes, int n_in,
                              void* d_out, int out_size, void* d_ws, size_t ws_size,
                              hipStream_t stream) {
  const float* pred     = (const float*)d_in[0];
  const float* targ     = (const float*)d_in[1];
  const int*   labelseq = (const int*)d_in[2];   // harness: integer -> const int*
  float*       out      = (float*)d_out;
  float*       partial  = (float*)d_ws;          // BLOCKS floats = 8 KB scratch

  dice_rows_kernel<<<BLOCKS, 256, 0, stream>>>(pred, targ, labelseq, partial);
  finalize_kernel<<<1, 256, 0, stream>>>(partial, out, BLOCKS);

  (void)in_sizes; (void)n_in; (void)out_size; (void)ws_size;
}